// Pairwise_Predictor_60722247630945
// MI455X (gfx1250) — compile-verified
//
#include <hip/hip_runtime.h>
#include <hip/hip_bf16.h>

// Pairwise predictor, MI455X (gfx1250, wave32).
// Phase 0: t = tanh(x)                      [64 x 1024]
// Phase 1: P = Wcat(256x64) x t (WMMA f32 16x16x4); rows 0..127 = pj,
//          rows 128..255 = pi + b1 (bias folded at store)
// Phase 2: e[i,j] = sum_h W2[h]*tanh(pi[h,i]+pj[h,j])   (compute-bound:
//          v_tanh_f32 on TRANS pipe co-executing with VALU FMAs; tiles
//          staged with GLOBAL_LOAD_ASYNC_TO_LDS_B128 / ASYNCcnt)
// Phase 3: out[i,j] = e[i,j] + e[j,i] + 2*b2

#define N_TOK 1024
#define C_IN  64
#define HID   128
#define TILE  32

typedef __attribute__((ext_vector_type(2))) float v2f;
typedef __attribute__((ext_vector_type(8))) float v8f;

__device__ __forceinline__ float fast_tanh(float x) {
#if __has_builtin(__builtin_amdgcn_tanhf)
  return __builtin_amdgcn_tanhf(x);   // v_tanh_f32 (gfx1250 TRANS op)
#else
  float r;
  asm("v_tanh_f32 %0, %1" : "=v"(r) : "v"(x));
  return r;
#endif
}

// Async global->LDS copy of 16 bytes (CDNA5 GLOBAL_LOAD_ASYNC_TO_LDS_B128,
// GV addressing mode; tracked with ASYNCcnt, no VGPR data round-trip).
__device__ __forceinline__ void async_copy_b128(void* lds_dst, const void* gsrc) {
  unsigned lds_addr = (unsigned)(unsigned long long)(unsigned long)(size_t)lds_dst;
  unsigned long long ga = (unsigned long long)(size_t)gsrc;
  asm volatile("global_load_async_to_lds_b128 %0, %1, off"
               :: "v"(lds_addr), "v"(ga) : "memory");
}

__device__ __forceinline__ void wait_asynccnt0() {
  asm volatile("s_wait_asynccnt 0x0" ::: "memory");
}

// ---------------- Phase 0: elementwise tanh of x ----------------
__global__ void tanh_x_kernel(const float* __restrict__ x,
                              float* __restrict__ t, int n) {
  int i = blockIdx.x * blockDim.x + threadIdx.x;
  if (i < n) t[i] = fast_tanh(x[i]);
}

// ---------------- Phase 1: P = Wcat x t via V_WMMA_F32_16X16X4_F32 --------
// Wcat[m,k] = (m < HID) ? W1[m, k] : W1[m-HID, C_IN + k]   (Wcat: 256 x 64)
// One wave computes one 16x16 tile of P (M=256, N=1024, K=64).
// A 16x4 f32 fragment: lanes 0-15 -> m=lane, {K=k0,k0+1}; lanes 16-31 -> {k0+2,k0+3}
// B 4x16 f32 fragment: mirrored (lanes 0-15 -> n=lane, rows k0/k0+1; 16-31 -> k0+2/k0+3)
// C/D: VGPR r -> (M = r + 8*(lane>=16), N = lane&15)
__global__ __launch_bounds__(128) void gemm_p_kernel(
    const float* __restrict__ W1, const float* __restrict__ b1,
    const float* __restrict__ t, float* __restrict__ P) {
  const int wave = threadIdx.x >> 5;
  const int lane = threadIdx.x & 31;
  const int tile = blockIdx.x * 4 + wave;   // 16 M-tiles x 64 N-tiles = 1024
  const int m0 = (tile >> 6) * 16;
  const int n0 = (tile & 63) * 16;
  const int half = lane >> 4;               // 0: K pair {0,1}, 1: K pair {2,3}
  const int lid  = lane & 15;
  const int m = m0 + lid;
  const int n = n0 + lid;

  const float* wrow = (m < HID) ? (W1 + m * (2 * C_IN))
                                : (W1 + (m - HID) * (2 * C_IN) + C_IN);
  v8f acc = {};
  for (int k0 = 0; k0 < C_IN; k0 += 4) {
    const int ks = k0 + half * 2;
    v2f a, b;
    a.x = wrow[ks];
    a.y = wrow[ks + 1];
    b.x = t[ks * N_TOK + n];
    b.y = t[(ks + 1) * N_TOK + n];
    acc = __builtin_amdgcn_wmma_f32_16x16x4_f32(
        /*neg_a=*/false, a, /*neg_b=*/false, b,
        /*c_mod=*/(short)0, acc, /*reuse_a=*/false, /*reuse_b=*/false);
  }
#pragma unroll
  for (int r = 0; r < 8; ++r) {
    const int row = m0 + r + half * 8;
    float v = acc[r];
    if (row >= HID) v += b1[row - HID];     // fold b1 into pi rows
    P[row * N_TOK + n0 + lid] = v;
  }
}

// ---------------- Phase 2: e[i,j] = sum_h W2[h]*tanh(pi[h,i]+pj[h,j]) -----
// 32x32 output tile per block (256 threads, 2x2 micro-tile/thread).
// pi/pj tile slabs async-staged in LDS: 2 * 128*32*4B = 32 KB (WGP has 320 KB).
__global__ __launch_bounds__(256) void pair_kernel(
    const float* __restrict__ P, const float* __restrict__ W2,
    float* __restrict__ e) {
  __shared__ float pi_s[HID * TILE];
  __shared__ float pj_s[HID * TILE];
  __shared__ float w2_s[HID];

  const int i0 = blockIdx.y * TILE;
  const int j0 = blockIdx.x * TILE;
  const float* pj_g = P;                 // rows 0..127
  const float* pi_g = P + HID * N_TOK;   // rows 128..255 (b1 already folded)

  // Async-stage tiles: 128 rows x 8 float4 per matrix, 256 threads -> 4 rounds,
  // 2 GLOBAL_LOAD_ASYNC_TO_LDS_B128 per thread per round (no VGPR round-trip).
  for (int r = threadIdx.x; r < HID * (TILE / 4); r += 256) {
    const int h = r >> 3;
    const int c = (r & 7) * 4;
    async_copy_b128(pi_s + h * TILE + c, pi_g + h * N_TOK + i0 + c);
    async_copy_b128(pj_s + h * TILE + c, pj_g + h * N_TOK + j0 + c);
  }
  if (threadIdx.x < HID) w2_s[threadIdx.x] = W2[threadIdx.x];
  wait_asynccnt0();
  __syncthreads();

  const int ti = (threadIdx.x & 15) * 2;
  const int tj = (threadIdx.x >> 4) * 2;
  float a00 = 0.f, a01 = 0.f, a10 = 0.f, a11 = 0.f;

#pragma unroll 4
  for (int h = 0; h < HID; ++h) {
    const float  w  = w2_s[h];
    const float2 pa = *(const float2*)(pi_s + h * TILE + ti);   // ds_load_b64
    const float2 pb = *(const float2*)(pj_s + h * TILE + tj);   // ds_load_b64
    a00 += w * fast_tanh(pa.x + pb.x);
    a01 += w * fast_tanh(pa.x + pb.y);
    a10 += w * fast_tanh(pa.y + pb.x);
    a11 += w * fast_tanh(pa.y + pb.y);
  }
  e[(i0 + ti)     * N_TOK + j0 + tj]     = a00;
  e[(i0 + ti)     * N_TOK + j0 + tj + 1] = a01;
  e[(i0 + ti + 1) * N_TOK + j0 + tj]     = a10;
  e[(i0 + ti + 1) * N_TOK + j0 + tj + 1] = a11;
}

// ---------------- Phase 3: out = e + e^T + 2*b2 ----------------
// e (4 MB) is L2-resident (192 MB L2), so transposed reads are cheap.
__global__ void symadd_kernel(const float* __restrict__ e,
                              const float* __restrict__ b2,
                              float* __restrict__ out) {
  const int idx = blockIdx.x * blockDim.x + threadIdx.x;
  const int i = idx >> 10;
  const int j = idx & (N_TOK - 1);
  out[idx] = e[idx] + e[j * N_TOK + i] + 2.0f * b2[0];
}

extern "C" void kernel_launch(void* const* d_in, const int* in_sizes, int n_in,
                              void* d_out, int out_size, void* d_ws, size_t ws_size,
                              hipStream_t stream) {
  const float* x  = (const float*)d_in[0];
  const float* W1 = (const float*)d_in[1];
  const float* b1 = (const float*)d_in[2];
  const float* W2 = (const float*)d_in[3];
  const float* b2 = (const float*)d_in[4];
  float* out = (float*)d_out;

  // Workspace layout (floats): t[64*1024] | P[256*1024] | e[1024*1024] = 5.25 MB
  float* t = (float*)d_ws;
  float* P = t + C_IN * N_TOK;
  float* e = P + 2 * HID * N_TOK;

  tanh_x_kernel<<<(C_IN * N_TOK) / 256, 256, 0, stream>>>(x, t, C_IN * N_TOK);
  gemm_p_kernel<<<256, 128, 0, stream>>>(W1, b1, t, P);
  pair_kernel<<<dim3(N_TOK / TILE, N_TOK / TILE), 256, 0, stream>>>(P, W2, e);
  symadd_kernel<<<(N_TOK * N_TOK) / 256, 256, 0, stream>>>(e, b2, out);
}